// DAM_4243427688539
// MI455X (gfx1250) — compile-verified
//
#include <hip/hip_runtime.h>
#include <hip/hip_bf16.h>
#include <math.h>

#define N_UNITS    1024
#define N_PATTERNS 4096

typedef __attribute__((ext_vector_type(2))) float v2f;
typedef __attribute__((ext_vector_type(8))) float v8f;

// ---------------------------------------------------------------------------
// Phase 1: W [4096 x 1024] row-major  ->  WT [1024 x 4096] row-major
// so that column i of W is a contiguous 16KB row of WT (coalesced streaming
// in the sequential phase; W fits in the 192MB L2 so re-reads are L2 hits).
// ---------------------------------------------------------------------------
__global__ void dam_transpose_kernel(const float* __restrict__ W,
                                     float* __restrict__ WT) {
    __shared__ float tile[32][33];
    const int col0 = blockIdx.x * 32;   // along n  (1024)
    const int row0 = blockIdx.y * 32;   // along K  (4096)
    const int tx = threadIdx.x;         // 0..31
    const int ty = threadIdx.y;         // 0..7
    #pragma unroll
    for (int r = 0; r < 32; r += 8)
        tile[ty + r][tx] = W[(size_t)(row0 + ty + r) * N_UNITS + col0 + tx];
    __syncthreads();
    #pragma unroll
    for (int r = 0; r < 32; r += 8)
        WT[(size_t)(col0 + ty + r) * N_PATTERNS + row0 + tx] = tile[tx][ty + r];
}

// ---------------------------------------------------------------------------
// Phase 2: h = W @ x via V_WMMA_F32_16X16X4_F32 (exact f32 math).
// One wave per 16-row tile of h (256 waves total). CDNA5 VGPR layouts:
//   A (16x4 f32): lanes 0-15 -> M=0..15 with {K=0,K=1}; lanes 16-31 -> {K=2,K=3}
//   B (4x16 f32): lane n (n=lane&15) holds column N=n; x lives in column N=0
//   D (16x16 f32): column N=0 -> lane 0 (M=0..7 in v[0..7]), lane 16 (M=8..15)
// B is built BRANCHLESSLY: every lane loads x[k+kOff] (broadcast addresses),
// then a cndmask zeroes all lanes except lane 0 / lane 16. The K-loop is then
// pure: load A, load B, 2 selects, wmma — no EXEC-mask control flow.
// ---------------------------------------------------------------------------
__global__ void dam_matvec_wmma_kernel(const float* __restrict__ W,
                                       const float* __restrict__ x,
                                       float* __restrict__ h) {
    const int lane    = threadIdx.x & 31;
    const int wave    = blockIdx.x * (blockDim.x >> 5) + (threadIdx.x >> 5); // 0..255
    const int rowBase = wave * 16;
    const int m       = lane & 15;
    const int kOff    = (lane >> 4) * 2;   // 0 for lanes 0-15, 2 for 16-31

    const float* Arow = W + (size_t)(rowBase + m) * N_UNITS + kOff;
    const float* Bsrc = x + kOff;
    const float  keep = (m == 0) ? 1.0f : 0.0f;   // lanes 0 and 16 carry column N=0

    v8f c = {};
    for (int k0 = 0; k0 < N_UNITS; k0 += 16) {
        #pragma unroll
        for (int u = 0; u < 4; ++u) {
            const int k = k0 + 4 * u;
            v2f a = *(const v2f*)(Arow + k);
            v2f bv = *(const v2f*)(Bsrc + k);
            v2f b;
            b.x = bv.x * keep;   // select-to-zero, branchless
            b.y = bv.y * keep;
            c = __builtin_amdgcn_wmma_f32_16x16x4_f32(
                    /*neg_a=*/false, a, /*neg_b=*/false, b,
                    /*c_mod=*/(short)0, c, /*reuse_a=*/false, /*reuse_b=*/false);
        }
    }
    if (lane == 0) {
        #pragma unroll
        for (int r = 0; r < 8; ++r) h[rowBase + r] = c[r];
    } else if (lane == 16) {
        #pragma unroll
        for (int r = 0; r < 8; ++r) h[rowBase + 8 + r] = c[r];
    }
}

// ---------------------------------------------------------------------------
// Phase 3: strict sequential sweep. One workgroup of 1024 threads (32 wave32s
// on one WGP). Thread t owns h[t], h[t+1024], h[t+2048], h[t+3072] in
// registers -> column loads from WT are fully coalesced. Per step:
//   diff = sum_k relu(h - 2 v c)^2 - relu(h)^2   (= e_pos - e_neg)
//   newv = tanh(diff);  h += (newv - v) * c
// Reduction: shfl_xor within wave, LDS across 32 waves, LDS broadcast back
// (fixed order -> deterministic). global_prefetch_b8 of the next 16KB column
// hides L2 latency behind the reduction + barriers.
// ---------------------------------------------------------------------------
__global__ void dam_sequential_kernel(const float* __restrict__ x,
                                      const float* __restrict__ WT,
                                      const float* __restrict__ h_in,
                                      float* __restrict__ out) {
    __shared__ float vals[N_UNITS];
    __shared__ float red[32];
    __shared__ float bcast;

    const int t    = threadIdx.x;        // 0..1023
    const int lane = t & 31;
    const int wid  = t >> 5;

    vals[t] = x[t];
    float h0 = h_in[t];
    float h1 = h_in[t + 1024];
    float h2 = h_in[t + 2048];
    float h3 = h_in[t + 3072];
    __syncthreads();

    for (int i = 0; i < N_UNITS; ++i) {
        const float* col = WT + (size_t)i * N_PATTERNS;
        const float c0 = col[t];
        const float c1 = col[t + 1024];
        const float c2 = col[t + 2048];
        const float c3 = col[t + 3072];

        if (i + 1 < N_UNITS) {
            const float* ncol = WT + (size_t)(i + 1) * N_PATTERNS;
            __builtin_prefetch(ncol + t,        0, 0);
            __builtin_prefetch(ncol + t + 1024, 0, 0);
            __builtin_prefetch(ncol + t + 2048, 0, 0);
            __builtin_prefetch(ncol + t + 3072, 0, 0);
        }

        const float v  = vals[i];
        const float tv = 2.0f * v;

        float p = 0.0f;
        {
            float a0 = fmaxf(h0 - tv * c0, 0.0f), b0 = fmaxf(h0, 0.0f);
            float a1 = fmaxf(h1 - tv * c1, 0.0f), b1 = fmaxf(h1, 0.0f);
            float a2 = fmaxf(h2 - tv * c2, 0.0f), b2 = fmaxf(h2, 0.0f);
            float a3 = fmaxf(h3 - tv * c3, 0.0f), b3 = fmaxf(h3, 0.0f);
            p += a0 * a0 - b0 * b0;
            p += a1 * a1 - b1 * b1;
            p += a2 * a2 - b2 * b2;
            p += a3 * a3 - b3 * b3;
        }

        // intra-wave reduction (wave32)
        #pragma unroll
        for (int s = 16; s > 0; s >>= 1) p += __shfl_xor(p, s, 32);
        if (lane == 0) red[wid] = p;
        __syncthreads();

        // cross-wave reduction by wave 0
        if (wid == 0) {
            float s = red[lane];
            #pragma unroll
            for (int sm = 16; sm > 0; sm >>= 1) s += __shfl_xor(s, sm, 32);
            if (lane == 0) {
                const float nv = tanhf(s);
                bcast   = nv;
                vals[i] = nv;
            }
        }
        __syncthreads();

        const float d = bcast - v;
        h0 = fmaf(d, c0, h0);
        h1 = fmaf(d, c1, h1);
        h2 = fmaf(d, c2, h2);
        h3 = fmaf(d, c3, h3);
    }

    out[t] = vals[t];
}

// ---------------------------------------------------------------------------
extern "C" void kernel_launch(void* const* d_in, const int* in_sizes, int n_in,
                              void* d_out, int out_size, void* d_ws, size_t ws_size,
                              hipStream_t stream) {
    const float* x = (const float*)d_in[0];           // [1024]
    const float* W = (const float*)d_in[1];           // [4096, 1024]
    float* out = (float*)d_out;                       // [1024]

    float* WT = (float*)d_ws;                         // [1024, 4096] = 16 MB
    float* h  = WT + (size_t)N_UNITS * N_PATTERNS;    // [4096]

    // Phase 1: transpose W into workspace
    dam_transpose_kernel<<<dim3(N_UNITS / 32, N_PATTERNS / 32),
                           dim3(32, 8), 0, stream>>>(W, WT);

    // Phase 2: h = W @ x via f32 WMMA (256 waves = 32 blocks x 8 waves)
    dam_matvec_wmma_kernel<<<32, 256, 0, stream>>>(W, x, h);

    // Phase 3: sequential DAM sweep, single 1024-thread workgroup
    dam_sequential_kernel<<<1, 1024, 0, stream>>>(x, WT, h, out);
}